// SPDenseTransLinear_17051020165439
// MI455X (gfx1250) — compile-verified
//
#include <hip/hip_runtime.h>
#include <stdint.h>

// Problem dims (match reference)
#define OUT_F 4096   // compressed rows
#define IN_F  2048   // compressed cols
#define BATCH 4096   // n
#define K_LOG 4096   // logical k = 2*IN_F
#define N_OUT 4096   // output columns = BATCH

typedef __attribute__((ext_vector_type(4)))  __bf16 v4bf;
typedef __attribute__((ext_vector_type(8)))  __bf16 v8bf;
typedef __attribute__((ext_vector_type(16))) __bf16 v16bf;
typedef __attribute__((ext_vector_type(32))) __bf16 v32bf;
typedef __attribute__((ext_vector_type(8)))  float  v8f;
typedef __attribute__((ext_vector_type(4)))  int    v4i;

#define AS1 __attribute__((address_space(1)))
#define AS3 __attribute__((address_space(3)))

// ---- CDNA5 async global->LDS copy (ASYNCcnt-tracked), with safe fallback ----
__device__ __forceinline__ void async_copy16(const void* g, void* l) {
#if __has_builtin(__builtin_amdgcn_global_load_async_to_lds_b128)
    __builtin_amdgcn_global_load_async_to_lds_b128(
        (AS1 v4i*)(uintptr_t)g,
        (AS3 v4i*)(uint32_t)(uintptr_t)l, 0, 0);
#else
    *(uint4*)l = *(const uint4*)g;   // fallback: global load + ds store
#endif
}

__device__ __forceinline__ void wait_async0() {
#if __has_builtin(__builtin_amdgcn_global_load_async_to_lds_b128)
#if __has_builtin(__builtin_amdgcn_s_wait_asynccnt)
    __builtin_amdgcn_s_wait_asynccnt(0);
#else
    asm volatile("s_wait_asynccnt 0x0" ::: "memory");
#endif
#endif
}

// ---------------- fp32 -> bf16 conversion (4 elems / thread) ----------------
__global__ void cvt_bf16_kernel(const float* __restrict__ src,
                                __bf16* __restrict__ dst, int n4) {
    int i = blockIdx.x * blockDim.x + threadIdx.x;
    if (i >= n4) return;
    const float4 f = ((const float4*)src)[i];
    v4bf o;
    o.x = (__bf16)f.x; o.y = (__bf16)f.y; o.z = (__bf16)f.z; o.w = (__bf16)f.w;
    ((v4bf*)dst)[i] = o;
}

// ------------- pack int32 metadata -> SWMMAC 2-bit index dwords -------------
// dword i covers metadata elements [16*i, 16*i+16): 8 groups, 4 bits each
// (idx0 in low 2 bits, idx1 in high 2 bits), group j at bit 4*j.
__global__ void pack_meta_kernel(const int* __restrict__ meta,
                                 uint32_t* __restrict__ idx, int ndw) {
    int i = blockIdx.x * blockDim.x + threadIdx.x;
    if (i >= ndw) return;
    const int* m = meta + (size_t)i * 16;
    uint32_t p = 0u;
#pragma unroll
    for (int j = 0; j < 8; ++j) {
        uint32_t i0 = (uint32_t)m[2 * j]     & 3u;
        uint32_t i1 = (uint32_t)m[2 * j + 1] & 3u;
        p |= (i0 | (i1 << 2)) << (4 * j);
    }
    idx[i] = p;
}

// ---------------------------- SWMMAC GEMM core ------------------------------
// Block: 256 threads = 8 waves (4 along M x 2 along N).
// Block tile: 128 compressed rows x 128 cols. Wave tile: 32 x 64.
// B tile (128 cols x 64 logical K bf16) staged in LDS via async copies,
// double-buffered; A (sparse, half-size) + index dwords stream from global.
#define BSTRIDE 72   // bf16 elems per LDS row (64 data + 8 pad = 144 B)

__global__ __launch_bounds__(256)
void spmm_swmmac_kernel(const __bf16* __restrict__ Wc,     // [OUT_F, IN_F] bf16
                        const __bf16* __restrict__ X,      // [BATCH, K_LOG] bf16
                        const uint32_t* __restrict__ Meta, // [OUT_F, IN_F/16]
                        const int* __restrict__ RowIdx,    // [OUT_F]
                        float* __restrict__ Out) {         // [2*OUT_F, N_OUT]
    __shared__ __bf16 Bsh[2][128 * BSTRIDE];               // 2 x 18 KB

    const int tid  = threadIdx.x;
    const int lane = tid & 31;
    const int wave = tid >> 5;
    const int wm   = wave & 3;   // 4 waves along M
    const int wn   = wave >> 2;  // 2 waves along N
    const int l16  = lane & 15;
    const int half = lane >> 4;

    const int r0   = blockIdx.y * 128 + wm * 32;  // compressed row base (wave)
    const int nblk = blockIdx.x * 128;            // output col base (block)

    const __bf16*   wp0 = Wc   + (size_t)(r0 + l16) * IN_F;
    const __bf16*   wp1 = wp0 + (size_t)16 * IN_F;
    const uint32_t* mp0 = Meta + (size_t)(r0 + l16) * (IN_F / 16);
    const uint32_t* mp1 = mp0 + (size_t)16 * (IN_F / 16);

    v8f acc[2][4] = {};

    // Stage a 128x64 B tile: 1024 chunks of 16 B; 4 chunks per thread.
    auto prefetch = [&](int buf, int kt) {
#pragma unroll
        for (int i = 0; i < 4; ++i) {
            const int c   = tid + 256 * i;   // 0..1023
            const int row = c >> 3;          // 0..127 (column of X tile)
            const int off = (c & 7) * 8;     // bf16 elems within 64-K row
            const __bf16* g = X + (size_t)(nblk + row) * K_LOG + kt * 64 + off;
            __bf16* l = &Bsh[buf][row * BSTRIDE + off];
            async_copy16(g, l);
        }
    };

    prefetch(0, 0);

    for (int kt = 0; kt < K_LOG / 64; ++kt) {
        const int cur = kt & 1;

        // A fragments (16x32 compressed bf16 each) straight from global.
        // lanes 0-15: c = ck0+0..7 (V0-3), ck0+16..23 (V4-7); lanes 16-31: +8.
        const int ck0 = kt * 32;
        v8bf a00 = *(const v8bf*)(wp0 + ck0 + half * 8);
        v8bf a01 = *(const v8bf*)(wp0 + ck0 + 16 + half * 8);
        v8bf a10 = *(const v8bf*)(wp1 + ck0 + half * 8);
        v8bf a11 = *(const v8bf*)(wp1 + ck0 + 16 + half * 8);
        v16bf A[2];
#pragma unroll
        for (int i = 0; i < 8; ++i) {
            A[0][i] = a00[i]; A[0][i + 8] = a01[i];
            A[1][i] = a10[i]; A[1][i + 8] = a11[i];
        }
        int sidx0 = (int)mp0[kt * 2 + half];
        int sidx1 = (int)mp1[kt * 2 + half];

        // Bsh[cur] ready: drain own ASYNCcnt, then workgroup barrier.
        wait_async0();
        __syncthreads();

        // All waves are past compute(kt-1), so Bsh[1-cur] is free to refill.
        if (kt + 1 < K_LOG / 64) prefetch(1 - cur, kt + 1);

        const __bf16* bbase = &Bsh[cur][0];
#pragma unroll
        for (int nt = 0; nt < 4; ++nt) {
            // B 64x16: lane holds col n (=lane%16); K run = half*16..+15 (V0-7)
            // and +32 (V8-15) within this 64-K stage.
            const __bf16* bp = bbase + (wn * 64 + nt * 16 + l16) * BSTRIDE + half * 16;
            v16bf b0 = *(const v16bf*)(bp);
            v16bf b1 = *(const v16bf*)(bp + 32);
            v32bf B;
#pragma unroll
            for (int i = 0; i < 16; ++i) { B[i] = b0[i]; B[i + 16] = b1[i]; }

            acc[0][nt] = __builtin_amdgcn_swmmac_f32_16x16x64_bf16(
                false, A[0], false, B, acc[0][nt], sidx0, false, false);
            acc[1][nt] = __builtin_amdgcn_swmmac_f32_16x16x64_bf16(
                false, A[1], false, B, acc[1][nt], sidx1, false, false);
        }
    }

    // Epilogue: scatter 16x16 C tiles to logical rows via RowIdx.
    // C layout: VGPR e -> M = e + half*8, N = lane%16.
#pragma unroll
    for (int mt = 0; mt < 2; ++mt) {
#pragma unroll
        for (int e = 0; e < 8; ++e) {
            const int lrow = RowIdx[r0 + mt * 16 + e + half * 8];
            float* op = Out + (size_t)lrow * N_OUT + nblk + wn * 64 + l16;
            op[0]  = acc[mt][0][e];
            op[16] = acc[mt][1][e];
            op[32] = acc[mt][2][e];
            op[48] = acc[mt][3][e];
        }
    }
}

// ------------------------------- launcher -----------------------------------
extern "C" void kernel_launch(void* const* d_in, const int* in_sizes, int n_in,
                              void* d_out, int out_size, void* d_ws, size_t ws_size,
                              hipStream_t stream) {
    const float* weight   = (const float*)d_in[0];  // [OUT_F, IN_F] f32
    const int*   indices  = (const int*)d_in[1];    // [OUT_F]
    const int*   metadata = (const int*)d_in[2];    // [OUT_F, IN_F]
    const float* input    = (const float*)d_in[3];  // [BATCH, K_LOG] f32
    float* out = (float*)d_out;                     // [2*OUT_F, N_OUT] f32

    // Workspace layout: Wbf 16 MB | Xbf 32 MB | Idx 2 MB  (total 50 MB)
    __bf16*   Wbf = (__bf16*)d_ws;
    __bf16*   Xbf = (__bf16*)((char*)d_ws + (size_t)16 * 1024 * 1024);
    uint32_t* Idx = (uint32_t*)((char*)d_ws + (size_t)48 * 1024 * 1024);

    // Zero output: only 4096 of 8192 logical rows get written by the GEMM.
    (void)hipMemsetAsync(d_out, 0, (size_t)out_size * sizeof(float), stream);

    {
        int n4 = OUT_F * IN_F / 4;
        cvt_bf16_kernel<<<(n4 + 255) / 256, 256, 0, stream>>>(weight, Wbf, n4);
    }
    {
        int n4 = BATCH * K_LOG / 4;
        cvt_bf16_kernel<<<(n4 + 255) / 256, 256, 0, stream>>>(input, Xbf, n4);
    }
    {
        int ndw = OUT_F * (IN_F / 16);
        pack_meta_kernel<<<(ndw + 255) / 256, 256, 0, stream>>>(metadata, Idx, ndw);
    }

    dim3 grid(N_OUT / 128, OUT_F / 128);
    spmm_swmmac_kernel<<<grid, 256, 0, stream>>>(Wbf, Xbf, Idx, indices, out);
}